// OptionHead_20899310862667
// MI455X (gfx1250) — compile-verified
//
#include <hip/hip_runtime.h>

typedef __attribute__((ext_vector_type(2))) float v2f;
typedef __attribute__((ext_vector_type(8))) float v8f;

#define NB   2048   // batch
#define NE   64     // experts
#define ND   512    // latent dim (K)
#define NO   512    // out dim (N)
#define MAXT 192    // max M-tiles: ceil(B/16) + E

// ---------------- prep kernels (build grouped-GEMM metadata) ----------------

__global__ void zero_meta_kernel(int* counts, int* cursors) {
    int t = blockIdx.x * blockDim.x + threadIdx.x;
    if (t < NE) { counts[t] = 0; cursors[t] = 0; }
}

__global__ void histo_kernel(const int* __restrict__ opt, int* counts) {
    int t = blockIdx.x * blockDim.x + threadIdx.x;
    if (t < NB) atomicAdd(&counts[opt[t]], 1);
}

// single-thread scan + tile-table build (64 experts -> trivial)
__global__ void scan_tiles_kernel(const int* __restrict__ counts, int* offsets,
                                  int* tileE, int* tileBase, int* tileCnt) {
    if (threadIdx.x != 0 || blockIdx.x != 0) return;
    int acc = 0;
    for (int e = 0; e < NE; ++e) { offsets[e] = acc; acc += counts[e]; }
    int t = 0;
    for (int e = 0; e < NE; ++e) {
        int c = counts[e];
        for (int i = 0; i < c; i += 16) {
            tileE[t]    = e;
            tileBase[t] = offsets[e] + i;
            tileCnt[t]  = (c - i) < 16 ? (c - i) : 16;
            ++t;
        }
    }
    for (; t < MAXT; ++t) { tileE[t] = -1; tileBase[t] = 0; tileCnt[t] = 0; }
}

__global__ void scatter_kernel(const int* __restrict__ opt,
                               const int* __restrict__ offsets,
                               int* cursors, int* idx) {
    int t = blockIdx.x * blockDim.x + threadIdx.x;
    if (t < NB) {
        int e = opt[t];
        int p = offsets[e] + atomicAdd(&cursors[e], 1);
        idx[p] = t;
    }
}

// ---------------- grouped GEMM: one 16-row sample tile per block ------------
// block = 256 threads = 8 wave32. Wave w computes TWO 16-col N-tiles
// (n and n+128), sharing one A operand per K-step -> 2 WMMAs per ds_load.
// V_WMMA_F32_16X16X4_F32 over K=512 (128 steps, 256 wmma issues per wave).

#define APAD 4   // row stride 516 floats: 16B-aligned rows, bank-conflict-free

__global__ void __launch_bounds__(256)
option_gemm_kernel(const float* __restrict__ x, const float* __restrict__ w,
                   const float* __restrict__ bias, const int* __restrict__ idx,
                   const int* __restrict__ tileE, const int* __restrict__ tileBase,
                   const int* __restrict__ tileCnt, float* __restrict__ out) {
    __shared__ float As[16][ND + APAD];
    __shared__ int   rowIdx[16];

    const int e = tileE[blockIdx.x];
    if (e < 0) return;                       // block-uniform: EXEC stays all-1s
    const int base = tileBase[blockIdx.x];
    const int cnt  = tileCnt[blockIdx.x];
    const int tid  = threadIdx.x;

    // --- stage gathered A tile (16 x 512 f32 = 32KB) + row indices to LDS ---
    if (tid < 16) rowIdx[tid] = (tid < cnt) ? idx[base + tid] : 0;
    {
        const int r  = tid >> 4;             // row 0..15
        const int cs = (tid & 15) * 32;      // 32-float segment
        if (r < cnt) {
            const float* src = x + (size_t)idx[base + r] * ND + cs;
            #pragma unroll
            for (int i = 0; i < 8; ++i) {
                float4 v = *(const float4*)(src + i * 4);
                *(float4*)&As[r][cs + i * 4] = v;
            }
        } else {
            #pragma unroll
            for (int i = 0; i < 8; ++i) {
                float4 z = {0.f, 0.f, 0.f, 0.f};
                *(float4*)&As[r][cs + i * 4] = z;
            }
        }
    }
    __syncthreads();

    const int lane = tid & 31;
    const int wv   = tid >> 5;
    const int n0   = blockIdx.y * 256 + wv * 16 + (lane & 15);  // first N-tile col
    const int kh   = (lane >> 4) * 2;        // K sub-offset per ISA A layout
    const int row  = lane & 15;              // A row for this lane

    const float* wp0 = w + (size_t)e * (ND * NO) + n0;
    const float* wp1 = wp0 + 128;            // second N-tile, +128 cols

    v8f acc0 = {};
    v8f acc1 = {};
    #pragma unroll 4
    for (int k0 = 0; k0 < ND; k0 += 4) {
        const int k = k0 + kh;
        v2f a;                                // A 16x4: lane row, K={k,k+1}
        a.x = As[row][k];
        a.y = As[row][k + 1];
        v2f b0, b1;                           // B 4x16: coalesced along N
        b0.x = wp0[(size_t)k * NO];
        b0.y = wp0[(size_t)(k + 1) * NO];
        b1.x = wp1[(size_t)k * NO];
        b1.y = wp1[(size_t)(k + 1) * NO];
        acc0 = __builtin_amdgcn_wmma_f32_16x16x4_f32(
            false, a, false, b0, (short)0, acc0, false, false);
        acc1 = __builtin_amdgcn_wmma_f32_16x16x4_f32(
            false, a, false, b1, (short)0, acc1, false, false);
    }

    // --- epilogue: bias add + scatter rows back (C/D layout: M=j | j+8) ---
    const float bv0 = bias[e * NO + n0];
    const float bv1 = bias[e * NO + n0 + 128];
    const int mhi = (lane >> 4) * 8;
    #pragma unroll
    for (int j = 0; j < 8; ++j) {
        const int m = j + mhi;
        if (m < cnt) {
            float* orow = out + (size_t)rowIdx[m] * NO + n0;
            orow[0]   = acc0[j] + bv0;
            orow[128] = acc1[j] + bv1;
        }
    }
}

// ---------------------------------------------------------------------------

extern "C" void kernel_launch(void* const* d_in, const int* in_sizes, int n_in,
                              void* d_out, int out_size, void* d_ws, size_t ws_size,
                              hipStream_t stream) {
    const float* x    = (const float*)d_in[0];   // [B, D]
    const int*   opt  = (const int*)d_in[1];     // [B]
    const float* wgt  = (const float*)d_in[2];   // [E, D, O]
    const float* bias = (const float*)d_in[3];   // [E, O]
    float* out = (float*)d_out;                  // [B, O]

    int* counts   = (int*)d_ws;          // 64
    int* offsets  = counts + NE;         // 64
    int* cursors  = offsets + NE;        // 64
    int* idx      = cursors + NE;        // 2048
    int* tileE    = idx + NB;            // 192
    int* tileBase = tileE + MAXT;        // 192
    int* tileCnt  = tileBase + MAXT;     // 192

    zero_meta_kernel<<<1, 64, 0, stream>>>(counts, cursors);
    histo_kernel<<<NB / 256, 256, 0, stream>>>(opt, counts);
    scan_tiles_kernel<<<1, 1, 0, stream>>>(counts, offsets, tileE, tileBase, tileCnt);
    scatter_kernel<<<NB / 256, 256, 0, stream>>>(opt, offsets, cursors, idx);

    dim3 grid(MAXT, 2);   // 192 M-tiles x 2 N-groups (8 waves x 2x16 cols each)
    option_gemm_kernel<<<grid, 256, 0, stream>>>(x, wgt, bias, idx,
                                                 tileE, tileBase, tileCnt, out);
}